// SoftQuantize_83262236000542
// MI455X (gfx1250) — compile-verified
//
#include <hip/hip_runtime.h>
#include <hip/hip_bf16.h>

typedef __attribute__((ext_vector_type(2))) float v2f;
typedef __attribute__((ext_vector_type(4))) float v4f;
typedef __attribute__((ext_vector_type(8))) float v8f;

#define L_CENTERS 256
#define M_DIM 4
#define NTILES (L_CENTERS / 16)
#define RECF 12   // floats per packed center record (48 B): conflict-free 16-lane stride

// ---- DPP-based 16-lane butterfly helpers (xor1/2/4 via DPP8, xor8 via row_ror:8) ----
constexpr int DPP8_SEL(int a, int b, int c, int d, int e, int f, int g, int h) {
    return a | (b << 3) | (c << 6) | (d << 9) | (e << 12) | (f << 15) | (g << 18) | (h << 21);
}
constexpr int SEL_XOR1 = DPP8_SEL(1, 0, 3, 2, 5, 4, 7, 6);
constexpr int SEL_XOR2 = DPP8_SEL(2, 3, 0, 1, 6, 7, 4, 5);
constexpr int SEL_XOR4 = DPP8_SEL(4, 5, 6, 7, 0, 1, 2, 3);

template <int SEL>
__device__ __forceinline__ float dpp8f(float x) {
    return __int_as_float(__builtin_amdgcn_mov_dpp8(__float_as_int(x), SEL));
}
__device__ __forceinline__ float dpp_ror8(float x) {
    // DPP16 row_ror:8 = lane ^ 8 within each 16-lane row
    return __int_as_float(__builtin_amdgcn_mov_dpp(__float_as_int(x), 0x128, 0xf, 0xf, true));
}
__device__ __forceinline__ float rsum16(float x) {
    x += dpp8f<SEL_XOR1>(x);
    x += dpp8f<SEL_XOR2>(x);
    x += dpp8f<SEL_XOR4>(x);
    x += dpp_ror8(x);
    return x;
}
__device__ __forceinline__ float rmax16(float x) {
    x = fmaxf(x, dpp8f<SEL_XOR1>(x));
    x = fmaxf(x, dpp8f<SEL_XOR2>(x));
    x = fmaxf(x, dpp8f<SEL_XOR4>(x));
    x = fmaxf(x, dpp_ror8(x));
    return x;
}

// Packed per-center record in LDS:
//   [0..3]  cs0..cs3 = 2*sig*log2e * c[m][l]   (WMMA B operand rows)
//   [4..7]  cw0..cw3 = c[m][l]                 (output weights)
//   [8]     bias     = -sig*log2e * ||c_l||^2  (WMMA C operand)
//   [9..11] pad

__global__ __launch_bounds__(256) void softquant_wmma_kernel(
    const float* __restrict__ x,
    const float* __restrict__ c,
    const float* __restrict__ sigma,
    float* __restrict__ out,
    int nvec)
{
    __shared__ float lPack[L_CENTERS * RECF];

    const float LOG2E = 1.4426950408889634f;
    const float sig = fmaxf(sigma[0], 0.0f) + 1e-4f;

    // ---- stage codebook into LDS: one center per thread (blockDim.x == 256) ----
    {
        const int l = threadIdx.x;
        const float c0 = c[0 * L_CENTERS + l];
        const float c1 = c[1 * L_CENTERS + l];
        const float c2 = c[2 * L_CENTERS + l];
        const float c3 = c[3 * L_CENTERS + l];
        const float s2 = 2.0f * sig * LOG2E;
        float* rec = &lPack[l * RECF];
        v4f cs; cs[0] = s2 * c0; cs[1] = s2 * c1; cs[2] = s2 * c2; cs[3] = s2 * c3;
        v4f cw; cw[0] = c0; cw[1] = c1; cw[2] = c2; cw[3] = c3;
        *reinterpret_cast<v4f*>(rec + 0) = cs;   // 16-B aligned (RECF*4 = 48)
        *reinterpret_cast<v4f*>(rec + 4) = cw;   // 16-B aligned
        rec[8] = -sig * LOG2E * (c0 * c0 + c1 * c1 + c2 * c2 + c3 * c3);
    }
    __syncthreads();

    const int lane = threadIdx.x & 31;
    const int half = lane >> 4;        // 0: lanes 0-15, 1: lanes 16-31
    const int col  = lane & 15;        // N index within tile / row index for A load
    const int waveId = blockIdx.x * (blockDim.x >> 5) + (threadIdx.x >> 5);
    const long vecbase = (long)waveId * 16;
    if (vecbase >= (long)nvec) return;

    // ---- A matrix: 16 vectors x 4 dims. Lane layout: v0 = K(2*half), v1 = K(2*half+1)
    v2f A;
    {
        const float* ap = x + (vecbase + col) * 4 + half * 2;
        A = *reinterpret_cast<const v2f*>(ap);   // global_load_b64, wave reads 256 B
    }

    // ---- score GEMM: D = A(16x4) * B(4x16) + bias, 16 tiles covering L=256 ----
    v8f D[NTILES];
    const int kb = half * 2;
#pragma unroll
    for (int t = 0; t < NTILES; ++t) {
        const int l = t * 16 + col;
        const float* rec = &lPack[l * RECF];
        const v2f Bm = *reinterpret_cast<const v2f*>(rec + kb);  // ds_load_b64
        const float bv = rec[8];                                 // ds_load_b32
        v8f C;
#pragma unroll
        for (int i = 0; i < 8; ++i) C[i] = bv;   // bias depends only on column N
        D[t] = __builtin_amdgcn_wmma_f32_16x16x4_f32(
            false, A, false, Bm, (short)0, C, false, false);
    }

    // ---- row max over all 256 centers (per-lane over tiles, then DPP butterfly) ----
    v8f mx = D[0];
#pragma unroll
    for (int t = 1; t < NTILES; ++t)
#pragma unroll
        for (int i = 0; i < 8; ++i) mx[i] = fmaxf(mx[i], D[t][i]);
#pragma unroll
    for (int i = 0; i < 8; ++i) mx[i] = rmax16(mx[i]);

    // ---- exp2, denominator, and 4 weighted numerators ----
    v8f den, n0, n1, n2, n3;
#pragma unroll
    for (int i = 0; i < 8; ++i) { den[i] = 0.f; n0[i] = 0.f; n1[i] = 0.f; n2[i] = 0.f; n3[i] = 0.f; }
#pragma unroll
    for (int t = 0; t < NTILES; ++t) {
        const int l = t * 16 + col;
        const v4f w = *reinterpret_cast<const v4f*>(&lPack[l * RECF] + 4);  // ds_load_b128
#pragma unroll
        for (int i = 0; i < 8; ++i) {
            const float e = __builtin_amdgcn_exp2f(D[t][i] - mx[i]);
            den[i] += e;
            n0[i] += e * w[0];
            n1[i] += e * w[1];
            n2[i] += e * w[2];
            n3[i] += e * w[3];
        }
    }

    // ---- DPP butterfly sums (halves hold disjoint row sets M=v / M=v+8) ----
#pragma unroll
    for (int i = 0; i < 8; ++i) {
        den[i] = rsum16(den[i]);
        n0[i]  = rsum16(n0[i]);
        n1[i]  = rsum16(n1[i]);
        n2[i]  = rsum16(n2[i]);
        n3[i]  = rsum16(n3[i]);
    }

    // ---- write: slot v in half h is vector row (v + 8h); lane col==v writes it ----
#pragma unroll
    for (int v = 0; v < 8; ++v) {
        if (col == v) {
            const long vec = vecbase + v + 8 * half;
            const float inv = 1.0f / den[v];
            float4 o;
            o.x = n0[v] * inv;
            o.y = n1[v] * inv;
            o.z = n2[v] * inv;
            o.w = n3[v] * inv;
            *reinterpret_cast<float4*>(out + vec * 4) = o;
        }
    }
}

extern "C" void kernel_launch(void* const* d_in, const int* in_sizes, int n_in,
                              void* d_out, int out_size, void* d_ws, size_t ws_size,
                              hipStream_t stream) {
    const float* x     = (const float*)d_in[0];   // (8192, 512) f32
    const float* c     = (const float*)d_in[1];   // (4, 256)    f32
    const float* sigma = (const float*)d_in[2];   // (1,)        f32
    float* out = (float*)d_out;                   // (8192, 512) f32

    const int nvec  = in_sizes[0] / M_DIM;        // 1,048,576 four-dim vectors
    const int waves = (nvec + 15) / 16;           // 16 vectors per wave
    const int wavesPerBlock = 8;                  // 256 threads, wave32
    const int blocks = (waves + wavesPerBlock - 1) / wavesPerBlock;

    softquant_wmma_kernel<<<blocks, 256, 0, stream>>>(x, c, sigma, out, nvec);
}